// GATConv_6150393168664
// MI455X (gfx1250) — compile-verified
//
#include <hip/hip_runtime.h>
#include <math.h>

#define NN 50000
#define EE 1600000
#define INC 128
#define HEADS 8
#define OUTC 16
#define HC 128            // HEADS*OUTC
#define NEG_SLOPE 0.2f

typedef __attribute__((ext_vector_type(2))) float v2f;
typedef __attribute__((ext_vector_type(8))) float v8f;

__device__ __forceinline__ float lrelu(float v) {
    return v > 0.0f ? v : NEG_SLOPE * v;
}
// order-preserving float<->int mapping so atomicMax(int) == float max
__device__ __forceinline__ int f2ord(float f) {
    int i = __float_as_int(f);
    return i >= 0 ? i : (i ^ 0x7fffffff);
}
__device__ __forceinline__ float ord2f(int i) {
    return __int_as_float(i >= 0 ? i : (i ^ 0x7fffffff));
}

// ---------------------------------------------------------------------------
// Kernel 1: h = x @ W via V_WMMA_F32_16X16X4_F32. One wave per 16x16 tile.
// Grid: 3125 blocks * 8 waves = 25000 tiles = (50000/16) * (128/16).
// ---------------------------------------------------------------------------
__global__ void __launch_bounds__(256) gat_gemm_wmma(
        const float* __restrict__ x, const float* __restrict__ W,
        float* __restrict__ h) {
    const int lane = threadIdx.x & 31;
    const int wave = threadIdx.x >> 5;
    const int tile = blockIdx.x * 8 + wave;
    const int tileM = tile >> 3;       // 0..3124
    const int tileN = tile & 7;        // 0..7 (one head per column tile)
    const int row0 = tileM * 16;
    const int col0 = tileN * 16;
    const int l16  = lane & 15;
    const int half = lane >> 4;        // 0: K0/K1 rows, 1: K2/K3 rows

    // A: lane holds x[row][k0 + 2*half + {0,1}]
    const float* xrow = x + (size_t)(row0 + l16) * INC + 2 * half;
    // B: lane holds W[k0 + 2*half + {0,1}][col0 + l16]
    const float* wcol = W + (size_t)(2 * half) * HC + col0 + l16;

    v8f acc = {};
#pragma unroll
    for (int k0 = 0; k0 < INC; k0 += 4) {
        v2f a = *(const v2f*)(xrow + k0);
        v2f b;
        b.x = wcol[(size_t)k0 * HC];
        b.y = wcol[(size_t)k0 * HC + HC];
        acc = __builtin_amdgcn_wmma_f32_16x16x4_f32(
            /*neg_a=*/false, a, /*neg_b=*/false, b,
            /*c_mod=*/(short)0, acc, /*reuse_a=*/false, /*reuse_b=*/false);
    }
    // D layout: lanes 0-15 rows row0+0..7, lanes 16-31 rows row0+8..15
    float* hp = h + (size_t)(row0 + half * 8) * HC + col0 + l16;
#pragma unroll
    for (int r = 0; r < 8; ++r) hp[(size_t)r * HC] = acc[r];
}

// ---------------------------------------------------------------------------
// Kernel 2: init out accumulator (zeros), segment-max (-inf ordinal), denom.
// ---------------------------------------------------------------------------
__global__ void gat_init(float* __restrict__ out, int* __restrict__ m_ord,
                         float* __restrict__ denom) {
    int idx = blockIdx.x * blockDim.x + threadIdx.x;
    if (idx < NN * HC) out[idx] = 0.0f;
    if (idx < NN * HEADS) {
        m_ord[idx] = f2ord(-INFINITY);
        denom[idx] = 0.0f;
    }
}

// ---------------------------------------------------------------------------
// Kernel 3: per (node,head) attention dot products a_src / a_dst.
// ---------------------------------------------------------------------------
__global__ void gat_node_att(const float* __restrict__ h,
                             const float* __restrict__ att_src,
                             const float* __restrict__ att_dst,
                             float* __restrict__ a_src,
                             float* __restrict__ a_dst) {
    int idx = blockIdx.x * blockDim.x + threadIdx.x;
    if (idx >= NN * HEADS) return;
    const int n  = idx >> 3;
    const int hd = idx & 7;
    const float4* hv = (const float4*)(h + (size_t)n * HC + hd * OUTC);
    const float4* as = (const float4*)(att_src + hd * OUTC);
    const float4* ad = (const float4*)(att_dst + hd * OUTC);
    float ss = 0.0f, dd = 0.0f;
#pragma unroll
    for (int i = 0; i < 4; ++i) {
        float4 v = hv[i], s = as[i], d = ad[i];
        ss += v.x * s.x + v.y * s.y + v.z * s.z + v.w * s.w;
        dd += v.x * d.x + v.y * d.y + v.z * d.z + v.w * d.w;
    }
    a_src[idx] = ss;
    a_dst[idx] = dd;
}

// ---------------------------------------------------------------------------
// Kernel 4: segment max of edge logits (per destination) via int atomicMax.
// ---------------------------------------------------------------------------
__global__ void gat_edge_max(const int* __restrict__ src,
                             const int* __restrict__ dst,
                             const float* __restrict__ a_src,
                             const float* __restrict__ a_dst,
                             int* __restrict__ m_ord) {
    int e = blockIdx.x * blockDim.x + threadIdx.x;
    if (e >= EE) return;
    const int s = src[e], d = dst[e];
    const float* asp = a_src + (size_t)s * HEADS;
    const float* adp = a_dst + (size_t)d * HEADS;
#pragma unroll
    for (int hd = 0; hd < HEADS; ++hd) {
        float ev = lrelu(asp[hd] + adp[hd]);
        atomicMax(m_ord + (size_t)d * HEADS + hd, f2ord(ev));
    }
}

// ---------------------------------------------------------------------------
// Kernel 5: decode max ordinal -> float, empty segments (-inf) -> 0.
// (in place: reuse the same buffer as float)
// ---------------------------------------------------------------------------
__global__ void gat_finalize_max(int* __restrict__ m_ord) {
    int idx = blockIdx.x * blockDim.x + threadIdx.x;
    if (idx >= NN * HEADS) return;
    float mv = ord2f(m_ord[idx]);
    if (!isfinite(mv)) mv = 0.0f;
    ((float*)m_ord)[idx] = mv;
}

// ---------------------------------------------------------------------------
// Kernel 6: denom = segment_sum(exp(e - m[dst]))
// ---------------------------------------------------------------------------
__global__ void gat_edge_denom(const int* __restrict__ src,
                               const int* __restrict__ dst,
                               const float* __restrict__ a_src,
                               const float* __restrict__ a_dst,
                               const float* __restrict__ m_f,
                               float* __restrict__ denom) {
    int e = blockIdx.x * blockDim.x + threadIdx.x;
    if (e >= EE) return;
    const int s = src[e], d = dst[e];
    const float* asp = a_src + (size_t)s * HEADS;
    const float* adp = a_dst + (size_t)d * HEADS;
    const float* mp  = m_f   + (size_t)d * HEADS;
#pragma unroll
    for (int hd = 0; hd < HEADS; ++hd) {
        float ev = lrelu(asp[hd] + adp[hd]);
        atomicAdd(denom + (size_t)d * HEADS + hd, expf(ev - mp[hd]));
    }
}

// ---------------------------------------------------------------------------
// Kernel 7: weighted scatter-add aggregation. Thread = (edge, head).
// ---------------------------------------------------------------------------
__global__ void gat_edge_aggregate(const int* __restrict__ src,
                                   const int* __restrict__ dst,
                                   const float* __restrict__ a_src,
                                   const float* __restrict__ a_dst,
                                   const float* __restrict__ m_f,
                                   const float* __restrict__ denom,
                                   const float* __restrict__ h,
                                   float* __restrict__ out) {
    int t = blockIdx.x * blockDim.x + threadIdx.x;
    if (t >= EE * HEADS) return;
    const int e  = t >> 3;
    const int hd = t & 7;
    const int s = src[e], d = dst[e];
    const int nh = d * HEADS + hd;
    float ev    = lrelu(a_src[(size_t)s * HEADS + hd] + a_dst[nh]);
    float ex    = expf(ev - m_f[nh]);
    float alpha = ex / (denom[nh] + 1e-16f);
    const float4* hv = (const float4*)(h + (size_t)s * HC + hd * OUTC);
    float* op = out + (size_t)d * HC + hd * OUTC;
#pragma unroll
    for (int i = 0; i < 4; ++i) {
        float4 v = hv[i];
        atomicAdd(op + 4 * i + 0, alpha * v.x);
        atomicAdd(op + 4 * i + 1, alpha * v.y);
        atomicAdd(op + 4 * i + 2, alpha * v.z);
        atomicAdd(op + 4 * i + 3, alpha * v.w);
    }
}

// ---------------------------------------------------------------------------
// Kernel 8: out = relu(out + bias)
// ---------------------------------------------------------------------------
__global__ void gat_bias_relu(float* __restrict__ out,
                              const float* __restrict__ bias) {
    int idx = blockIdx.x * blockDim.x + threadIdx.x;
    if (idx >= NN * HC) return;
    out[idx] = fmaxf(out[idx] + bias[idx & (HC - 1)], 0.0f);
}

extern "C" void kernel_launch(void* const* d_in, const int* in_sizes, int n_in,
                              void* d_out, int out_size, void* d_ws, size_t ws_size,
                              hipStream_t stream) {
    const float* x       = (const float*)d_in[0];
    // d_in[1] = x_0 (unused by reference)
    const float* W       = (const float*)d_in[2];
    const float* att_src = (const float*)d_in[3];
    const float* att_dst = (const float*)d_in[4];
    const float* bias    = (const float*)d_in[5];
    const int*   eidx    = (const int*)d_in[6];
    const int*   src     = eidx;
    const int*   dst     = eidx + EE;
    float* out = (float*)d_out;

    // workspace layout (floats): h | a_src | a_dst | m | denom  (~32 MB)
    float* h      = (float*)d_ws;
    float* a_src  = h + (size_t)NN * HC;
    float* a_dst  = a_src + (size_t)NN * HEADS;
    int*   m_ord  = (int*)(a_dst + (size_t)NN * HEADS);
    float* m_f    = (float*)m_ord;                      // reused in place
    float* denom  = (float*)(m_ord + (size_t)NN * HEADS);

    const int B = 256;
    // 1) h = x @ W (WMMA fp32)
    gat_gemm_wmma<<<(NN / 16) * (HC / 16) / 8, B, 0, stream>>>(x, W, h);
    // 2) init accumulators
    gat_init<<<(NN * HC + B - 1) / B, B, 0, stream>>>(out, m_ord, denom);
    // 3) node attention terms
    gat_node_att<<<(NN * HEADS + B - 1) / B, B, 0, stream>>>(h, att_src, att_dst, a_src, a_dst);
    // 4) segment max
    gat_edge_max<<<(EE + B - 1) / B, B, 0, stream>>>(src, dst, a_src, a_dst, m_ord);
    // 5) decode max
    gat_finalize_max<<<(NN * HEADS + B - 1) / B, B, 0, stream>>>(m_ord);
    // 6) softmax denominator
    gat_edge_denom<<<(EE + B - 1) / B, B, 0, stream>>>(src, dst, a_src, a_dst, m_f, denom);
    // 7) weighted aggregation
    gat_edge_aggregate<<<(EE * HEADS + B - 1) / B, B, 0, stream>>>(src, dst, a_src, a_dst,
                                                                   m_f, denom, h, out);
    // 8) bias + relu
    gat_bias_relu<<<(NN * HC + B - 1) / B, B, 0, stream>>>(out, bias);
}